// _WrappedSearcher_33294586479019
// MI455X (gfx1250) — compile-verified
//
#include <hip/hip_runtime.h>

// Problem constants (match reference)
constexpr int Bv = 512;     // queries
constexpr int Dv = 256;     // embedding dim
constexpr int Nv = 500000;  // database size
constexpr int Kv = 100;     // top-k

// Tiling
constexpr int M_TILE = 128;              // rows per block
constexpr int N_TILE = 64;               // cols per n-iteration
constexpr int K_STAGE = 64;              // K per LDS stage
constexpr int XS_STRIDE = Dv + 4;        // 260 : stride % 64 == 4 -> conflict-free frag reads
constexpr int KBS_STRIDE = K_STAGE + 4;  // 68  : stride % 64 == 4
constexpr int SS_STRIDE = N_TILE + 2;    // 66  : spreads row-halves across banks
constexpr int MAX_CHUNKS = 128;

typedef float v2f __attribute__((ext_vector_type(2)));
typedef float v8f __attribute__((ext_vector_type(8)));

// ---------------------------------------------------------------------------
// Phase 1: fp32 WMMA GEMM tile + streaming per-row top-K within a column chunk
// grid: (C chunks, 4 row-blocks), block: 256 threads (8 waves)
// ---------------------------------------------------------------------------
__global__ __launch_bounds__(256, 1) void topk_phase1(
    const float* __restrict__ x, const float* __restrict__ kb,
    float* __restrict__ cval, int* __restrict__ cidx,
    int C, int chunk_size)
{
  __shared__ float xs[M_TILE * XS_STRIDE];    // 133120 B : x tile, full K resident
  __shared__ float kbs[N_TILE * KBS_STRIDE];  //  17408 B : kb K-stage
  __shared__ float ss[M_TILE * SS_STRIDE];    //  33792 B : score tile
  __shared__ float tval[M_TILE][Kv];          //  51200 B : per-row top-K values
  __shared__ int   tidxs[M_TILE][Kv];         //  51200 B : per-row top-K indices

  const int tid  = threadIdx.x;
  const int wave = tid >> 5;
  const int lane = tid & 31;
  const int l16  = lane & 15;
  const int lhi  = lane >> 4;               // 0 or 1 (lane half)
  const int chunk = blockIdx.x;
  const int m0    = blockIdx.y * M_TILE;

  const int n_start = chunk * chunk_size;
  int n_end = n_start + chunk_size;
  if (n_end > Nv) n_end = Nv;

  // ---- load x tile (128 rows x 256 K) coalesced as float4 ----
  #pragma unroll
  for (int j = 0; j < 32; ++j) {
    int q   = j * 256 + tid;       // float4 id, 8192 total
    int row = q >> 6;              // 64 float4 per row
    int kk  = (q & 63) << 2;
    float4 v = *(const float4*)&x[(size_t)(m0 + row) * Dv + kk];
    *(float4*)&xs[row * XS_STRIDE + kk] = v;
  }
  __syncthreads();

  // per-row selection state (registers; only tid < M_TILE meaningful)
  int   cnt  = 0;
  int   mpos = 0;
  float thr  = -__builtin_inff();

  // fragment base pointers (A from xs, B from kbs); stride%64==4 => no bank conflicts
  const float* xsbase = &xs[(wave * 16 + l16) * XS_STRIDE + 2 * lhi];
  const float* kbbase = &kbs[l16 * KBS_STRIDE + 2 * lhi];

  float4 ld[4];  // register prefetch buffer: 64 cols x 64 K = 1024 float4 / 256 thr

  for (int n0 = n_start; n0 < n_end; n0 += N_TILE) {
    v8f acc[4];
    #pragma unroll
    for (int c = 0; c < 4; ++c) acc[c] = {};

    // prologue: issue global loads for stage 0
    #pragma unroll
    for (int j = 0; j < 4; ++j) {
      int q = j * 256 + tid;
      int col = q >> 4;                 // 16 float4 per col-row
      int kk  = (q & 15) << 2;
      int g = n0 + col;
      ld[j] = (g < n_end) ? *(const float4*)&kb[(size_t)g * Dv + kk]
                          : make_float4(0.f, 0.f, 0.f, 0.f);
    }

    for (int s = 0; s < Dv / K_STAGE; ++s) {
      __syncthreads();                  // kbs consumed by previous stage
      #pragma unroll
      for (int j = 0; j < 4; ++j) {
        int q = j * 256 + tid;
        int col = q >> 4;
        int kk  = (q & 15) << 2;
        *(float4*)&kbs[col * KBS_STRIDE + kk] = ld[j];
      }
      __syncthreads();

      // issue next stage's global loads; they fly while we do WMMAs
      if (s + 1 < Dv / K_STAGE) {
        const int k0n = (s + 1) * K_STAGE;
        #pragma unroll
        for (int j = 0; j < 4; ++j) {
          int q = j * 256 + tid;
          int col = q >> 4;
          int kk  = (q & 15) << 2;
          int g = n0 + col;
          ld[j] = (g < n_end) ? *(const float4*)&kb[(size_t)g * Dv + k0n + kk]
                              : make_float4(0.f, 0.f, 0.f, 0.f);
        }
      }

      // compute: wave owns row-tile `wave`, all 4 col-tiles; 64 WMMAs per stage
      const int k0 = s * K_STAGE;
      #pragma unroll
      for (int ks = 0; ks < K_STAGE / 4; ++ks) {
        v2f a = *(const v2f*)(xsbase + k0 + 4 * ks);   // A: 16x4 f32 fragment
        #pragma unroll
        for (int c = 0; c < 4; ++c) {
          v2f b = *(const v2f*)(kbbase + c * 16 * KBS_STRIDE + 4 * ks);  // B: 4x16
          acc[c] = __builtin_amdgcn_wmma_f32_16x16x4_f32(
              false, a, false, b, (short)0, acc[c], false, false);
        }
      }
    }

    // spill 128x64 score tile to LDS: lane holds rows i+8*lhi, col = ct*16+l16
    #pragma unroll
    for (int c = 0; c < 4; ++c) {
      #pragma unroll
      for (int i = 0; i < 8; ++i) {
        ss[(wave * 16 + i + 8 * lhi) * SS_STRIDE + c * 16 + l16] = acc[c][i];
      }
    }
    __syncthreads();

    // streaming selection: thread r owns row r; threshold filter + replace-min
    if (tid < M_TILE) {
      float* tv = tval[tid];
      int*   ti = tidxs[tid];
      const float* srow = &ss[tid * SS_STRIDE];
      int ncols = n_end - n0;
      if (ncols > N_TILE) ncols = N_TILE;
      for (int c2 = 0; c2 < ncols; ++c2) {
        float sv = srow[c2];
        if (cnt < Kv) {
          tv[cnt] = sv; ti[cnt] = n0 + c2; ++cnt;
          if (cnt == Kv) {
            thr = tv[0]; mpos = 0;
            for (int j = 1; j < Kv; ++j) if (tv[j] < thr) { thr = tv[j]; mpos = j; }
          }
        } else if (sv > thr) {
          tv[mpos] = sv; ti[mpos] = n0 + c2;
          thr = tv[0]; mpos = 0;
          for (int j = 1; j < Kv; ++j) if (tv[j] < thr) { thr = tv[j]; mpos = j; }
        }
      }
    }
    __syncthreads();
  }

  // write per-(row, chunk) candidates
  if (tid < M_TILE) {
    int rg = m0 + tid;
    size_t base = ((size_t)rg * C + chunk) * Kv;
    for (int j = 0; j < Kv; ++j) {
      bool ok = j < cnt;
      cval[base + j] = ok ? tval[tid][j] : -__builtin_inff();
      cidx[base + j] = ok ? tidxs[tid][j] : 0;
    }
  }
}

// ---------------------------------------------------------------------------
// Phase 2: per row merge C*K candidates -> sorted top-100 indices.
// Key = orderable(f32) << 32 | (~index): descending sort gives top_k tie order.
// ---------------------------------------------------------------------------
__global__ __launch_bounds__(256, 1) void topk_phase2(
    const float* __restrict__ cval, const int* __restrict__ cidx,
    int* __restrict__ out, int C)
{
  __shared__ unsigned long long key[MAX_CHUNKS * Kv <= 16384 ? 16384 : 16384];
  const int tid = threadIdx.x;
  const int row = blockIdx.x;
  const int ncand = C * Kv;
  int P = 1;
  while (P < ncand) P <<= 1;           // <= 16384 for C <= 128

  size_t base = (size_t)row * ncand;
  for (int i = tid; i < P; i += 256) {
    unsigned long long k;
    if (i < ncand) {
      unsigned int u = __float_as_uint(cval[base + i]);
      u = (u & 0x80000000u) ? ~u : (u | 0x80000000u);   // order-preserving map
      unsigned int id = (unsigned int)cidx[base + i];
      k = ((unsigned long long)u << 32) | (unsigned long long)(0xFFFFFFFFu - id);
    } else {
      k = 0ull;                         // sorts last
    }
    key[i] = k;
  }
  __syncthreads();

  // bitonic sort, descending
  for (int kk = 2; kk <= P; kk <<= 1) {
    for (int j = kk >> 1; j > 0; j >>= 1) {
      for (int i = tid; i < P; i += 256) {
        int ixj = i ^ j;
        if (ixj > i) {
          unsigned long long a = key[i], b = key[ixj];
          bool desc = ((i & kk) == 0);
          if (desc ? (a < b) : (a > b)) { key[i] = b; key[ixj] = a; }
        }
      }
      __syncthreads();
    }
  }

  for (int i = tid; i < Kv; i += 256) {
    out[(size_t)row * Kv + i] =
        (int)(0xFFFFFFFFu - (unsigned int)(key[i] & 0xFFFFFFFFull));
  }
}

// ---------------------------------------------------------------------------
extern "C" void kernel_launch(void* const* d_in, const int* in_sizes, int n_in,
                              void* d_out, int out_size, void* d_ws, size_t ws_size,
                              hipStream_t stream) {
  (void)in_sizes; (void)n_in; (void)out_size;
  const float* x  = (const float*)d_in[0];
  const float* kb = (const float*)d_in[1];
  int* out = (int*)d_out;

  // chunk count bounded by workspace: per chunk = 512 rows * 100 * (4+4) bytes
  const size_t per_chunk = (size_t)Bv * Kv * 8;
  int C = (int)(ws_size / per_chunk);
  if (C < 1) C = 1;
  if (C > MAX_CHUNKS) C = MAX_CHUNKS;
  int chunk_size = (Nv + C - 1) / C;

  float* cval = (float*)d_ws;
  int*   cidx = (int*)((char*)d_ws + (size_t)Bv * C * Kv * sizeof(float));

  topk_phase1<<<dim3(C, Bv / M_TILE), 256, 0, stream>>>(x, kb, cval, cidx, C, chunk_size);
  topk_phase2<<<dim3(Bv), 256, 0, stream>>>(cval, cidx, out, C);
}